// DiffeomorphicTransform2D_90125593740087
// MI455X (gfx1250) — compile-verified
//
#include <hip/hip_runtime.h>
#include <math.h>

#define WID 512
#define HEI 512
#define NPIX (WID*HEI)
#define NB 8
#define NCH 4
#define TILE 32
#define HALO 8
#define WIN (TILE + 2*HALO)   // 48
#define NUM_STEPS 7

// ---------------------------------------------------------------------------
// CDNA5 async global->LDS copy (8 bytes per lane), tracked by ASYNCcnt.
// VDST = 32-bit wave-relative LDS byte address, VADDR = 64-bit global address.
// ---------------------------------------------------------------------------
__device__ __forceinline__ void async_copy_b64(const void* gptr, void* lptr) {
    unsigned lds = (unsigned)(size_t)lptr;  // low 32 bits of generic shared ptr = LDS byte addr
    asm volatile("global_load_async_to_lds_b64 %0, %1, off"
                 :: "v"(lds), "v"(gptr) : "memory");
}

__device__ __forceinline__ void wait_async0() {
#if __has_builtin(__builtin_amdgcn_s_wait_asynccnt)
    __builtin_amdgcn_s_wait_asynccnt(0);
#else
    asm volatile("s_wait_asynccnt 0" ::: "memory");
#endif
}

// ---------------------------------------------------------------------------
// flow0 = velocity / 2^NUM_STEPS, planar [B,2,H,W] -> interleaved float2(x,y)
// ---------------------------------------------------------------------------
__global__ __launch_bounds__(256) void k_init(const float* __restrict__ vf,
                                              float2* __restrict__ f0, int total) {
    int t = blockIdx.x * 256 + threadIdx.x;
    if (t >= total) return;
    int b = t / NPIX, p = t - b * NPIX;
    const float s = 1.0f / 128.0f;            // 1 / 2^7
    float fy = vf[(size_t)(b * 2 + 0) * NPIX + p] * s;  // channel 0 = y
    float fx = vf[(size_t)(b * 2 + 1) * NPIX + p] * s;  // channel 1 = x
    f0[t] = make_float2(fx, fy);
}

// ---------------------------------------------------------------------------
// One scaling-and-squaring step: fout(p) = fin(p) + bilinear(fin, p + fin(p))
// 32x32 tile + 8px halo staged in LDS via async global->LDS copies.
// ---------------------------------------------------------------------------
__global__ __launch_bounds__(256) void k_step(const float2* __restrict__ fin,
                                              float2* __restrict__ fout) {
    __shared__ float2 win[WIN * WIN];         // 18432 bytes
    const int tx = threadIdx.x;
    const int tilesPerRow = WID / TILE;       // 16
    const int tilesPerImg = tilesPerRow * (HEI / TILE);  // 256
    const int b  = blockIdx.x / tilesPerImg;
    const int tr = blockIdx.x - b * tilesPerImg;
    const int x0 = (tr % tilesPerRow) * TILE;
    const int y0 = (tr / tilesPerRow) * TILE;
    const int wx0 = x0 - HALO, wy0 = y0 - HALO;
    const float2* fb = fin + (size_t)b * NPIX;

    // Stage window (coords clamped -> legal addresses; clamped halo cells are
    // never read because the zero-padding test runs before the window test).
#pragma unroll
    for (int i = 0; i < (WIN * WIN) / 256; ++i) {   // 2304/256 = 9, no tail
        int s  = tx + i * 256;
        int sx = wx0 + (s % WIN);
        int sy = wy0 + (s / WIN);
        sx = min(max(sx, 0), WID - 1);
        sy = min(max(sy, 0), HEI - 1);
        async_copy_b64(fb + (sy * WID + sx), &win[s]);
    }
    wait_async0();
    __syncthreads();

    const int lx  = tx & 31;
    const int ly0 = tx >> 5;
    const float SC = (float)WID / (float)(WID - 1);  // 512/511

#pragma unroll
    for (int k = 0; k < 4; ++k) {
        int ly = ly0 + k * 8;
        int px = x0 + lx, py = y0 + ly;
        float2 f = win[(ly + HALO) * WIN + (lx + HALO)];  // fin at own pixel
        float ix = ((float)px + f.x) * SC - 0.5f;
        float iy = ((float)py + f.y) * SC - 0.5f;
        float fx0 = floorf(ix), fy0 = floorf(iy);
        int ix0 = (int)fx0, iy0 = (int)fy0;
        float wx1 = ix - fx0, wy1 = iy - fy0;

        float ax = 0.0f, ay = 0.0f;
#pragma unroll
        for (int dy = 0; dy < 2; ++dy) {
            int jy = iy0 + dy;
            float wy = dy ? wy1 : (1.0f - wy1);
            if (jy < 0 || jy >= HEI) continue;       // zeros padding
#pragma unroll
            for (int dx = 0; dx < 2; ++dx) {
                int jx = ix0 + dx;
                if (jx < 0 || jx >= WID) continue;   // zeros padding
                float wgt = wy * (dx ? wx1 : (1.0f - wx1));
                int rx = jx - wx0, ry = jy - wy0;
                float2 v;
                if ((unsigned)rx < (unsigned)WIN && (unsigned)ry < (unsigned)WIN)
                    v = win[ry * WIN + rx];          // LDS fast path
                else
                    v = fb[jy * WID + jx];           // rare outlier: L2 gather
                ax += wgt * v.x;
                ay += wgt * v.y;
            }
        }
        fout[(size_t)b * NPIX + py * WID + px] = make_float2(f.x + ax, f.y + ay);
    }
}

// ---------------------------------------------------------------------------
// Final warp: out[b,c] = bilinear(src[b,c], grid + flow), planar 4-channel.
// ---------------------------------------------------------------------------
__global__ __launch_bounds__(256) void k_warp(const float* __restrict__ src,
                                              const float2* __restrict__ flow,
                                              float* __restrict__ out, int total) {
    int t = blockIdx.x * 256 + threadIdx.x;
    if (t >= total) return;
    int b = t / NPIX, p = t - b * NPIX;
    int py = p >> 9, px = p & (WID - 1);
    float2 f = flow[t];
    const float SC = (float)WID / (float)(WID - 1);
    float ix = ((float)px + f.x) * SC - 0.5f;
    float iy = ((float)py + f.y) * SC - 0.5f;
    float fx0 = floorf(ix), fy0 = floorf(iy);
    int ix0 = (int)fx0, iy0 = (int)fy0;
    float wx1 = ix - fx0, wy1 = iy - fy0;

    const float* sb = src + (size_t)b * NCH * NPIX;
    float a0 = 0.0f, a1 = 0.0f, a2 = 0.0f, a3 = 0.0f;
#pragma unroll
    for (int dy = 0; dy < 2; ++dy) {
        int jy = iy0 + dy;
        float wy = dy ? wy1 : (1.0f - wy1);
        if (jy < 0 || jy >= HEI) continue;
#pragma unroll
        for (int dx = 0; dx < 2; ++dx) {
            int jx = ix0 + dx;
            if (jx < 0 || jx >= WID) continue;
            float wgt = wy * (dx ? wx1 : (1.0f - wx1));
            int o = jy * WID + jx;
            a0 += wgt * sb[o];
            a1 += wgt * sb[o + NPIX];
            a2 += wgt * sb[o + 2 * NPIX];
            a3 += wgt * sb[o + 3 * NPIX];
        }
    }
    size_t ob = (size_t)b * NCH * NPIX + p;
    out[ob]            = a0;
    out[ob + NPIX]     = a1;
    out[ob + 2 * NPIX] = a2;
    out[ob + 3 * NPIX] = a3;
}

// ---------------------------------------------------------------------------
extern "C" void kernel_launch(void* const* d_in, const int* in_sizes, int n_in,
                              void* d_out, int out_size, void* d_ws, size_t ws_size,
                              hipStream_t stream) {
    const float* src = (const float*)d_in[0];   // [8,4,512,512]
    const float* vf  = (const float*)d_in[1];   // [8,2,512,512]
    float* out = (float*)d_out;                 // [8,4,512,512]

    float2* w0 = (float2*)d_ws;                 // 16 MB
    float2* w1 = w0 + (size_t)NB * NPIX;        // 16 MB

    const int total = NB * NPIX;
    k_init<<<(total + 255) / 256, 256, 0, stream>>>(vf, w0, total);

    float2* a = w0;
    float2* b = w1;
    const int stepBlocks = NB * (WID / TILE) * (HEI / TILE);  // 2048
    for (int i = 0; i < NUM_STEPS; ++i) {
        k_step<<<stepBlocks, 256, 0, stream>>>(a, b);
        float2* tmp = a; a = b; b = tmp;        // after 7 steps, result in w1
    }

    k_warp<<<(total + 255) / 256, 256, 0, stream>>>(src, a, out, total);
}